// AttentiveDecoder_86861418594534
// MI455X (gfx1250) — compile-verified
//
#include <hip/hip_runtime.h>
#include <hip/hip_bf16.h>
#include <math.h>

// ---------------------------------------------------------------------------
// Problem constants: B=64, L=2048, H=E=512, V=32000
// ---------------------------------------------------------------------------
#define BB   64
#define LL   2048
#define HH   512
#define VV   32000

typedef __attribute__((ext_vector_type(2))) float v2f;
typedef __attribute__((ext_vector_type(4))) float v4f;
typedef __attribute__((ext_vector_type(8))) float v8f;

// ---------------------------------------------------------------------------
// Wave-level fp32 WMMA GEMM:  C[64 x N] = act( A[64 x K] * W[N x K]^T + bias )
// 4 waves/block; each wave owns one 16-wide N tile and all four 16-row
// M-tiles (weights streamed once, each B fragment feeds 4 WMMAs).
//
// K is processed in 8-wide blocks with ONE b128 load per fragment per block:
// lane (half h) loads floats [kb+4h .. kb+4h+3]. WMMA lets the fragment's
// K=0..3 slots map to any 4 global k's as long as A and B agree, so:
//   step 0 uses f.xy  -> global k { kb,kb+1, kb+4,kb+5 }
//   step 1 uses f.zw  -> global k { kb+2,kb+3, kb+6,kb+7 }
// => 5 b128 loads + 8 WMMAs per 8-k block, depth-1 software pipeline.
//
// V_WMMA_F32_16X16X4_F32 C layout: vgpr r -> row half*8+r, col = lane&15.
// ---------------------------------------------------------------------------
__global__ void wmma_gemm_m64(const float* __restrict__ A, int lda,
                              const float* __restrict__ W,   // N x K, row major
                              const float* __restrict__ bias,
                              float* __restrict__ C, int ldc,
                              int K, int act /*0=none,1=tanh*/)
{
    const int wave = threadIdx.x >> 5;
    const int lane = threadIdx.x & 31;
    const int n0   = (blockIdx.x * 4 + wave) << 4;   // 16-wide N tile
    const int half = lane >> 4;                       // 0 / 1
    const int lid  = lane & 15;

    v8f acc[4];
#pragma unroll
    for (int mt = 0; mt < 4; ++mt) acc[mt] = {};

    // 32-bit element offsets from uniform bases (all buffers < 4 GB)
    const unsigned woff = (unsigned)(n0 + lid) * (unsigned)K + 4u * half;
    unsigned aoff[4];
#pragma unroll
    for (int mt = 0; mt < 4; ++mt)
        aoff[mt] = (unsigned)(mt * 16 + lid) * (unsigned)lda + 4u * half;

    // prologue: fragments for kb = 0
    v4f bq = *(const v4f*)(W + woff);
    v4f aq[4];
#pragma unroll
    for (int mt = 0; mt < 4; ++mt) aq[mt] = *(const v4f*)(A + aoff[mt]);

    // steady state: prefetch block kb, compute block kb-8
    for (int kb = 8; kb < K; kb += 8) {
        v4f bq_n = *(const v4f*)(W + woff + (unsigned)kb);
        v4f aq_n[4];
#pragma unroll
        for (int mt = 0; mt < 4; ++mt)
            aq_n[mt] = *(const v4f*)(A + aoff[mt] + (unsigned)kb);

        v2f b0 = __builtin_shufflevector(bq, bq, 0, 1);
        v2f b1 = __builtin_shufflevector(bq, bq, 2, 3);
#pragma unroll
        for (int mt = 0; mt < 4; ++mt) {
            v2f a0 = __builtin_shufflevector(aq[mt], aq[mt], 0, 1);
            acc[mt] = __builtin_amdgcn_wmma_f32_16x16x4_f32(
                false, a0, false, b0, (short)0, acc[mt], false, false);
        }
#pragma unroll
        for (int mt = 0; mt < 4; ++mt) {
            v2f a1 = __builtin_shufflevector(aq[mt], aq[mt], 2, 3);
            acc[mt] = __builtin_amdgcn_wmma_f32_16x16x4_f32(
                false, a1, false, b1, (short)0, acc[mt], false, false);
        }
        bq = bq_n;
#pragma unroll
        for (int mt = 0; mt < 4; ++mt) aq[mt] = aq_n[mt];
    }

    // epilogue: last 8-k block
    {
        v2f b0 = __builtin_shufflevector(bq, bq, 0, 1);
        v2f b1 = __builtin_shufflevector(bq, bq, 2, 3);
#pragma unroll
        for (int mt = 0; mt < 4; ++mt) {
            v2f a0 = __builtin_shufflevector(aq[mt], aq[mt], 0, 1);
            acc[mt] = __builtin_amdgcn_wmma_f32_16x16x4_f32(
                false, a0, false, b0, (short)0, acc[mt], false, false);
        }
#pragma unroll
        for (int mt = 0; mt < 4; ++mt) {
            v2f a1 = __builtin_shufflevector(aq[mt], aq[mt], 2, 3);
            acc[mt] = __builtin_amdgcn_wmma_f32_16x16x4_f32(
                false, a1, false, b1, (short)0, acc[mt], false, false);
        }
    }

    const float bn = bias ? bias[n0 + lid] : 0.0f;
#pragma unroll
    for (int mt = 0; mt < 4; ++mt) {
#pragma unroll
        for (int r = 0; r < 8; ++r) {
            float v = acc[mt][r] + bn;
            if (act) v = tanhf(v);
            C[(size_t)(mt * 16 + half * 8 + r) * ldc + n0 + lid] = v;
        }
    }
}

// ---------------------------------------------------------------------------
// Embedding gather + input_feeding passthrough copy
// ---------------------------------------------------------------------------
__global__ void gather_copy(const int* __restrict__ seq,
                            const float* __restrict__ emb,
                            float* __restrict__ x,
                            const float* __restrict__ feed,
                            float* __restrict__ feed_out)
{
    int i = blockIdx.x * blockDim.x + threadIdx.x;     // 0 .. 64*512-1
    int b = i >> 9;
    int e = i & 511;
    x[i]        = emb[(size_t)seq[b] * HH + e];
    feed_out[i] = feed[i];
}

// ---------------------------------------------------------------------------
// GRU gate fusion: h_new from gi/gh (64x1536 each) and previous h.
// Writes h_new into ws_concat[:, 0:512] and into d_out last_hidden region.
// ---------------------------------------------------------------------------
__global__ void gru_gate(const float* __restrict__ gi,
                         const float* __restrict__ gh,
                         const float* __restrict__ hprev,   // 64x512
                         float* __restrict__ concat_buf,    // 64x1024 (ld 1024)
                         float* __restrict__ hid_out)       // 64x512
{
    int i = blockIdx.x * blockDim.x + threadIdx.x;     // 0 .. 64*512-1
    int b = i >> 9;
    int j = i & 511;
    size_t g = (size_t)b * 1536 + j;
    float ir = gi[g],        hr = gh[g];
    float iz = gi[g + 512],  hz = gh[g + 512];
    float in = gi[g + 1024], hn = gh[g + 1024];
    float r  = 1.0f / (1.0f + __expf(-(ir + hr)));
    float z  = 1.0f / (1.0f + __expf(-(iz + hz)));
    float n  = tanhf(in + r * hn);
    float hv = (1.0f - z) * n + z * hprev[i];
    concat_buf[(size_t)b * 1024 + j] = hv;
    hid_out[i] = hv;
}

// ---------------------------------------------------------------------------
// Fused attention: one pass over encoder_outputs (268 MB read ONCE).
// Block = 256 threads (8 waves) per batch row. Each wave runs an online
// softmax over 256 keys, holding its 512-float context accumulator in
// registers (16 floats/lane); partial (max,sum,acc) merged through LDS.
// Raw scores + (gmax,gsum) stashed for the attn_weights epilogue.
// ---------------------------------------------------------------------------
__global__ void attn_fused(const float* __restrict__ concat_buf, // h_new at ld 1024
                           const float* __restrict__ enc,        // B,L,H
                           float* __restrict__ scores,           // B,L raw dots
                           float* __restrict__ msum,             // B*2: gmax,gsum
                           float* __restrict__ ctx_out)          // concat_buf col 512+
{
    __shared__ float h_s[HH];
    __shared__ float red[8 * HH];     // 16 KB partial contexts
    __shared__ float wm[8], wsum[8];

    const int b    = blockIdx.x;
    const int t    = threadIdx.x;
    const int wave = t >> 5;
    const int lane = t & 31;

    for (int i = t; i < HH; i += 256) h_s[i] = concat_buf[(size_t)b * 1024 + i];
    __syncthreads();

    // per-lane cache of h_new[b]: h index = c*128 + lane*4 + j
    float hr[16];
#pragma unroll
    for (int c = 0; c < 4; ++c) {
        float4 v = *(const float4*)(h_s + c * 128 + lane * 4);
        hr[c * 4 + 0] = v.x; hr[c * 4 + 1] = v.y;
        hr[c * 4 + 2] = v.z; hr[c * 4 + 3] = v.w;
    }

    float m = -__builtin_inff();
    float s = 0.0f;
    float acc[16];
#pragma unroll
    for (int i = 0; i < 16; ++i) acc[i] = 0.0f;

    const float* encb = enc + (size_t)b * LL * HH;
    const int l0 = wave * (LL / 8), l1 = l0 + (LL / 8);
    for (int l = l0; l < l1; ++l) {
        const float* row = encb + (size_t)l * HH;
        float ev[16];
        float p = 0.0f;
#pragma unroll
        for (int c = 0; c < 4; ++c) {
            float4 v = *(const float4*)(row + c * 128 + lane * 4);
            ev[c * 4 + 0] = v.x; ev[c * 4 + 1] = v.y;
            ev[c * 4 + 2] = v.z; ev[c * 4 + 3] = v.w;
            p += v.x * hr[c * 4 + 0] + v.y * hr[c * 4 + 1]
               + v.z * hr[c * 4 + 2] + v.w * hr[c * 4 + 3];
        }
#pragma unroll
        for (int off = 16; off > 0; off >>= 1) p += __shfl_xor(p, off, 32);
        if (lane == 0) scores[(size_t)b * LL + l] = p;

        float mn   = fmaxf(m, p);
        float corr = __expf(m - mn);
        float e    = __expf(p - mn);
        s = s * corr + e;
#pragma unroll
        for (int i = 0; i < 16; ++i) acc[i] = acc[i] * corr + e * ev[i];
        m = mn;
    }

    if (lane == 0) { wm[wave] = m; wsum[wave] = s; }
#pragma unroll
    for (int c = 0; c < 4; ++c) {
#pragma unroll
        for (int j = 0; j < 4; ++j)
            red[wave * HH + c * 128 + lane * 4 + j] = acc[c * 4 + j];
    }
    __syncthreads();

    float gm = wm[0];
#pragma unroll
    for (int w = 1; w < 8; ++w) gm = fmaxf(gm, wm[w]);
    float gs = 0.0f;
#pragma unroll
    for (int w = 0; w < 8; ++w) gs += wsum[w] * __expf(wm[w] - gm);
    if (t == 0) { msum[b * 2] = gm; msum[b * 2 + 1] = gs; }

    for (int h = t; h < HH; h += 256) {
        float a = 0.0f;
#pragma unroll
        for (int w = 0; w < 8; ++w) a += red[w * HH + h] * __expf(wm[w] - gm);
        ctx_out[(size_t)b * 1024 + 512 + h] = a / gs;
    }
}

// ---------------------------------------------------------------------------
// attn_weights epilogue: attn = exp(score - gmax)/gsum
// ---------------------------------------------------------------------------
__global__ void attn_norm(const float* __restrict__ scores,
                          const float* __restrict__ msum,
                          float* __restrict__ attn_out)
{
    int i = blockIdx.x * blockDim.x + threadIdx.x;   // 0 .. 64*2048-1
    int b = i >> 11;
    attn_out[i] = __expf(scores[i] - msum[b * 2]) / msum[b * 2 + 1];
}

// ---------------------------------------------------------------------------
// Launch
// ---------------------------------------------------------------------------
extern "C" void kernel_launch(void* const* d_in, const int* in_sizes, int n_in,
                              void* d_out, int out_size, void* d_ws, size_t ws_size,
                              hipStream_t stream)
{
    const int*   seq      = (const int*)  d_in[0];
    const float* hidden   = (const float*)d_in[1];   // (1,64,512)
    const float* enc      = (const float*)d_in[2];   // (64,2048,512)
    /* d_in[3] encoder_labels unused by the reference */
    const float* feed     = (const float*)d_in[4];   // (64,512)
    const float* emb      = (const float*)d_in[5];   // (32000,512)
    const float* W_ih     = (const float*)d_in[6];   // (1536,512)
    const float* W_hh     = (const float*)d_in[7];   // (1536,512)
    const float* b_ih     = (const float*)d_in[8];
    const float* b_hh     = (const float*)d_in[9];
    const float* W_concat = (const float*)d_in[10];  // (512,1024)
    const float* b_concat = (const float*)d_in[11];
    const float* W_out    = (const float*)d_in[12];  // (32000,512)
    const float* b_out    = (const float*)d_in[13];

    float* out      = (float*)d_out;                 // (64,32000)
    float* hid_out  = out + (size_t)BB * VV;         // (1,64,512)
    float* attn_out = hid_out + (size_t)BB * HH;     // (64,1,2048)
    float* feed_out = attn_out + (size_t)BB * LL;    // (64,512)

    float* ws       = (float*)d_ws;
    float* ws_x     = ws;                            // 64x512
    float* ws_gi    = ws_x    + (size_t)BB * HH;     // 64x1536
    float* ws_gh    = ws_gi   + (size_t)BB * 1536;   // 64x1536
    float* ws_cat   = ws_gh   + (size_t)BB * 1536;   // 64x1024  [h_new | ctx]
    float* ws_sc    = ws_cat  + (size_t)BB * 1024;   // 64x2048 raw scores
    float* ws_ms    = ws_sc   + (size_t)BB * LL;     // 64x2 (gmax,gsum)
    float* ws_co    = ws_ms   + (size_t)BB * 2;      // 64x512 concat_output

    // 1) embedding gather + input_feeding passthrough
    gather_copy<<<BB * HH / 256, 256, 0, stream>>>(seq, emb, ws_x, feed, feed_out);

    // 2,3) GRU input/hidden gate GEMMs (fp32 WMMA), 4 N-tiles per block
    wmma_gemm_m64<<<1536 / 64, 128, 0, stream>>>(ws_x,   HH, W_ih, b_ih, ws_gi, 1536, HH, 0);
    wmma_gemm_m64<<<1536 / 64, 128, 0, stream>>>(hidden, HH, W_hh, b_hh, ws_gh, 1536, HH, 0);

    // 4) gate nonlinearity -> h_new (into concat buffer + hidden output)
    gru_gate<<<BB * HH / 256, 256, 0, stream>>>(ws_gi, ws_gh, hidden, ws_cat, hid_out);

    // 5) fused attention: encoder_outputs streamed exactly once
    attn_fused<<<BB, 256, 0, stream>>>(ws_cat, enc, ws_sc, ws_ms, ws_cat);

    // 6) attn_weights output
    attn_norm<<<BB * LL / 256, 256, 0, stream>>>(ws_sc, ws_ms, attn_out);

    // 7) concat projection with tanh (K=1024)
    wmma_gemm_m64<<<HH / 64, 128, 0, stream>>>(ws_cat, 1024, W_concat, b_concat, ws_co, HH, 1024, 1);

    // 8) vocab projection (dominant GEMM, 500 blocks x 4 waves)
    wmma_gemm_m64<<<VV / 64, 128, 0, stream>>>(ws_co, HH, W_out, b_out, out, VV, HH, 0);
}